// DrafterAttention_2662879724218
// MI455X (gfx1250) — compile-verified
//
#include <hip/hip_runtime.h>
#include <math.h>

// Problem constants (from reference)
#define B_    8
#define H_    16
#define D_    128
#define S_    4096
#define HID_  2048
#define EPS_  1e-6f
#define SCALE_ 0.08838834764831845f   // 1/sqrt(128)

typedef float v2f __attribute__((ext_vector_type(2)));
typedef float v4f __attribute__((ext_vector_type(4)));
typedef float v8f __attribute__((ext_vector_type(8)));

// ---------------------------------------------------------------------------
// Kernel 1/4: C[8 x 2048] = A[8 x 2048] * W[2048 x 2048] using
// V_WMMA_F32_16X16X4_F32. One block per 16-column tile (128 blocks),
// 8 waves per block splitting K, LDS reduction of the 8-VGPR accumulators.
// A row index is clamped (m & 7): rows 8..15 of D are discarded at the
// epilogue, so no zero-padding (and no EXEC divergence) is needed.
// 4 independent accumulators break the WMMA RAW chain.
// ---------------------------------------------------------------------------
#define GM_WAVES 8
#define GM_THREADS (GM_WAVES * 32)

__global__ __launch_bounds__(GM_THREADS)
void gemm8x2048_wmma(const float* __restrict__ Amat,
                     const float* __restrict__ W,
                     float* __restrict__ Out) {
  const int n0   = blockIdx.x * 16;
  const int lane = threadIdx.x & 31;
  const int wave = threadIdx.x >> 5;
  const int half = lane >> 4;     // 0: lanes 0-15, 1: lanes 16-31
  const int l15  = lane & 15;
  const int m    = l15 & 7;       // clamped A row (rows 8..15 of D discarded)

  __shared__ float red[GM_WAVES][32][8];

  const int kBeg = wave * (HID_ / GM_WAVES);
  const int kEnd = kBeg + (HID_ / GM_WAVES);

  const float* arow = Amat + (size_t)m * HID_ + half * 2;
  const float* bcol = W + (size_t)(half * 2) * HID_ + n0 + l15;

  auto step = [&](int kk, v8f cin) -> v8f {
    // A tile 16x4 (MxK), 32-bit layout: v0 = K(0|2), v1 = K(1|3)
    v2f a;
    a.x = arow[kk];
    a.y = arow[kk + 1];
    // B tile 4x16 (KxN): v0 = rows kk+(0|2), v1 = rows kk+(1|3), col = n0+l15
    const float* bp = bcol + (size_t)kk * HID_;
    v2f b;
    b.x = bp[0];
    b.y = bp[HID_];
    return __builtin_amdgcn_wmma_f32_16x16x4_f32(false, a, false, b,
                                                 (short)0, cin, false, false);
  };

  v8f c0 = {}, c1 = {}, c2 = {}, c3 = {};
  for (int k = kBeg; k < kEnd; k += 16) {
    c0 = step(k + 0,  c0);
    c1 = step(k + 4,  c1);
    c2 = step(k + 8,  c2);
    c3 = step(k + 12, c3);
  }
  const v8f c = (c0 + c1) + (c2 + c3);

  // cross-wave reduction of partial accumulators
#pragma unroll
  for (int i = 0; i < 8; ++i) red[wave][lane][i] = c[i];
  __syncthreads();

  if (wave == 0 && half == 0) {
    // C/D layout: VGPR i, lanes 0-15 -> row M=i, col=lane. Rows 0..7 valid.
#pragma unroll
    for (int i = 0; i < 8; ++i) {
      float s = 0.0f;
#pragma unroll
      for (int w = 0; w < GM_WAVES; ++w) s += red[w][lane][i];
      Out[(size_t)i * HID_ + n0 + l15] = s;
    }
  }
}

// ---------------------------------------------------------------------------
// Kernel 2: per-(b,h) RMSNorm over D=128 + RoPE. 128 blocks x 128 threads.
// ---------------------------------------------------------------------------
__global__ __launch_bounds__(128)
void rmsnorm_rope(const float* __restrict__ q_lin,
                  const float* __restrict__ cosv,
                  const float* __restrict__ sinv,
                  const float* __restrict__ gamma,
                  float* __restrict__ q_rot) {
  const int bh = blockIdx.x;          // b*H + h
  const int t  = threadIdx.x;         // 0..127
  __shared__ float qn[D_];
  __shared__ float red[D_];

  float v = q_lin[(size_t)bh * D_ + t];   // q_lin[b][h*D + t] == [bh*D + t]
  red[t] = v * v;
  __syncthreads();
  for (int s = 64; s > 0; s >>= 1) {
    if (t < s) red[t] += red[t + s];
    __syncthreads();
  }
  const float rms = rsqrtf(red[0] * (1.0f / (float)D_) + EPS_);
  qn[t] = v * rms * gamma[t];
  __syncthreads();

  float* out = q_rot + (size_t)bh * D_;
  if (t < 64) {
    const float cs = cosv[t], sn = sinv[t];
    const float q1 = qn[t], q2 = qn[t + 64];
    out[t]      = q1 * cs - q2 * sn;
    out[t + 64] = q2 * cs + q1 * sn;
  }
}

// ---------------------------------------------------------------------------
// Kernel 3: streaming attention, one block per (b,h). 16 waves.
//   phase A: scores[s] = SCALE * <q, K[s,:]> + mask  (one wave per K row:
//            32 lanes x float4 == one contiguous 128-float row)
//   phase B: block softmax over 4096 scores in LDS
//   phase C: out[d] = <probs, V[d,:]>  (V row contiguous in s)
// KV loads are non-temporal: the 512 MB stream must not evict L2-resident
// weights (Wq/Wo/x fit comfortably in the 192 MB L2).
// ---------------------------------------------------------------------------
#define AT_THREADS 512
#define AT_WAVES   16

__global__ __launch_bounds__(AT_THREADS)
void attn_kernel(const float* __restrict__ q_rot,
                 const float* __restrict__ Kc,
                 const float* __restrict__ Vc,
                 const float* __restrict__ mask,
                 float* __restrict__ attn_out) {
  const int bh   = blockIdx.x;
  const int b    = bh / H_;
  const int tid  = threadIdx.x;
  const int lane = tid & 31;
  const int wave = tid >> 5;

  __shared__ float qs[D_];
  __shared__ float sc[S_];
  __shared__ float wred[AT_WAVES];

  if (tid < D_) qs[tid] = q_rot[(size_t)bh * D_ + tid];
  __syncthreads();

  // ---- phase A: scores -------------------------------------------------
  const float* Kb = Kc + (size_t)bh * S_ * D_;
  const v4f qv = *reinterpret_cast<const v4f*>(&qs[lane * 4]);
  for (int s = wave; s < S_; s += AT_WAVES) {
    const v4f kv = __builtin_nontemporal_load(
        reinterpret_cast<const v4f*>(Kb + (size_t)s * D_ + lane * 4));
    float p = kv.x * qv.x + kv.y * qv.y + kv.z * qv.z + kv.w * qv.w;
#pragma unroll
    for (int off = 16; off > 0; off >>= 1) p += __shfl_xor(p, off, 32);
    if (lane == 0) sc[s] = p * SCALE_ + mask[(size_t)b * S_ + s];
  }
  __syncthreads();

  // ---- phase B: softmax ------------------------------------------------
  const int base = tid * (S_ / AT_THREADS);   // 8 elements per thread
  float lmax = -INFINITY;
#pragma unroll
  for (int i = 0; i < S_ / AT_THREADS; ++i) lmax = fmaxf(lmax, sc[base + i]);
#pragma unroll
  for (int off = 16; off > 0; off >>= 1) lmax = fmaxf(lmax, __shfl_xor(lmax, off, 32));
  if (lane == 0) wred[wave] = lmax;
  __syncthreads();
  if (tid == 0) {
    float m = wred[0];
    for (int w = 1; w < AT_WAVES; ++w) m = fmaxf(m, wred[w]);
    wred[0] = m;
  }
  __syncthreads();
  const float gmax = wred[0];
  __syncthreads();

  float lsum = 0.0f;
#pragma unroll
  for (int i = 0; i < S_ / AT_THREADS; ++i) {
    float e = __expf(sc[base + i] - gmax);
    sc[base + i] = e;
    lsum += e;
  }
#pragma unroll
  for (int off = 16; off > 0; off >>= 1) lsum += __shfl_xor(lsum, off, 32);
  if (lane == 0) wred[wave] = lsum;
  __syncthreads();
  if (tid == 0) {
    float s = 0.0f;
    for (int w = 0; w < AT_WAVES; ++w) s += wred[w];
    wred[0] = 1.0f / s;
  }
  __syncthreads();
  const float inv = wred[0];
#pragma unroll
  for (int i = 0; i < S_ / AT_THREADS; ++i) sc[base + i] *= inv;
  __syncthreads();

  // ---- phase C: probs @ V^T -------------------------------------------
  const float* Vb = Vc + (size_t)bh * D_ * S_;
  for (int d = wave; d < D_; d += AT_WAVES) {
    const float* vrow = Vb + (size_t)d * S_;
    float acc = 0.0f;
    for (int s0 = lane * 4; s0 < S_; s0 += 32 * 4) {
      const v4f vv = __builtin_nontemporal_load(
          reinterpret_cast<const v4f*>(vrow + s0));
      acc += vv.x * sc[s0] + vv.y * sc[s0 + 1] +
             vv.z * sc[s0 + 2] + vv.w * sc[s0 + 3];
    }
#pragma unroll
    for (int off = 16; off > 0; off >>= 1) acc += __shfl_xor(acc, off, 32);
    if (lane == 0) attn_out[(size_t)bh * D_ + d] = acc;  // == [b][h*D+d]
  }
}

// ---------------------------------------------------------------------------
extern "C" void kernel_launch(void* const* d_in, const int* in_sizes, int n_in,
                              void* d_out, int out_size, void* d_ws, size_t ws_size,
                              hipStream_t stream) {
  (void)in_sizes; (void)n_in; (void)out_size; (void)ws_size;
  const float* x     = (const float*)d_in[0];
  const float* kv_k  = (const float*)d_in[1];
  const float* kv_v  = (const float*)d_in[2];
  const float* cosv  = (const float*)d_in[3];
  const float* sinv  = (const float*)d_in[4];
  const float* mask  = (const float*)d_in[5];
  const float* Wq    = (const float*)d_in[6];
  const float* Wo    = (const float*)d_in[7];
  const float* gamma = (const float*)d_in[8];
  float* out = (float*)d_out;

  float* q_lin  = (float*)d_ws;             // 8*2048 f32
  float* q_rot  = q_lin + B_ * HID_;        // 8*2048 f32
  float* attn_o = q_rot + B_ * HID_;        // 8*2048 f32

  gemm8x2048_wmma<<<HID_ / 16, GM_THREADS, 0, stream>>>(x, Wq, q_lin);
  rmsnorm_rope<<<B_ * H_, D_, 0, stream>>>(q_lin, cosv, sinv, gamma, q_rot);
  attn_kernel<<<B_ * H_, AT_THREADS, 0, stream>>>(q_rot, kv_k, kv_v, mask, attn_o);
  gemm8x2048_wmma<<<HID_ / 16, GM_THREADS, 0, stream>>>(attn_o, Wo, out);
}